// GCN2_82575041232956
// MI455X (gfx1250) — compile-verified
//
#include <hip/hip_runtime.h>
#include <hip/hip_bf16.h>

#define N_NODES 100000
#define N_EDGES 800000
#define D       128
#define BN_EPS  1e-5f

typedef float v2f __attribute__((ext_vector_type(2)));
typedef float v8f __attribute__((ext_vector_type(8)));

// LDS layout for W: K-pair interleaved. Pair p = k/2 holds {W[2p][c], W[2p+1][c]}
// contiguously at dword address p*PSTRIDE + 2c. PSTRIDE = 288 dwords (256+32 pad)
// => lanes 0-15 (pair p) hit banks 0-31, lanes 16-31 (pair p+1, +288 % 64 = 32)
// hit banks 32-63: conflict-free ds_load_b64 B-fragment reads.
#define PSTRIDE 288
#define W_LDS_BYTES (64 * PSTRIDE * 4)   // 73,728 B (dynamic LDS, < 320 KB WGP)

// ---------------------------------------------------------------------------
// Kernel 0: init agg=0, deg=1 (self-loop), column sums = 0
// ---------------------------------------------------------------------------
__global__ void gcn_init_kernel(float* __restrict__ agg, float* __restrict__ deg,
                                float* __restrict__ colsum, float* __restrict__ colsumsq) {
    size_t i = (size_t)blockIdx.x * 256 + threadIdx.x;
    if (i < (size_t)N_NODES * D) agg[i] = 0.0f;
    if (i < N_NODES)             deg[i] = 1.0f;     // self-loop contributes 1 to degree
    if (i < D) { colsum[i] = 0.0f; colsumsq[i] = 0.0f; }
}

// ---------------------------------------------------------------------------
// Kernel 1: degree accumulation over edges (dst side)
// ---------------------------------------------------------------------------
__global__ void gcn_deg_kernel(const int* __restrict__ dst, float* __restrict__ deg) {
    int e = blockIdx.x * 256 + threadIdx.x;
    if (e < N_EDGES) atomicAdd(&deg[dst[e]], 1.0f);
}

// ---------------------------------------------------------------------------
// Kernel 2: deg -> deg^{-1/2} in place
// ---------------------------------------------------------------------------
__global__ void gcn_rsqrt_kernel(float* __restrict__ deg) {
    int i = blockIdx.x * 256 + threadIdx.x;
    if (i < N_NODES) deg[i] = rsqrtf(deg[i]);
}

// ---------------------------------------------------------------------------
// Kernel 3: h = x @ W via V_WMMA_F32_16X16X4_F32 (fp32 exact, wave32)
//   Block = 256 threads = 8 waves; each wave owns a 16-row x 128-col tile.
//   W staged K-pair-interleaved in LDS so every B fragment is one ds_load_b64.
//   A frag (16x4):  lane<16 -> K=k0,k0+1 ; lane>=16 -> K=k0+2,k0+3  (row = lane&15)
//   B frag (4x16):  VGPR v, half h -> K = k0 + 2h + v, col = nt*16 + (lane&15)
//   C/D (16x16 f32): VGPR i, half h -> M = 8h + i, N = lane&15
// ---------------------------------------------------------------------------
__global__ void gcn_gemm_wmma_kernel(const float* __restrict__ x,
                                     const float* __restrict__ W,
                                     float* __restrict__ h) {
    extern __shared__ float wlds[];            // 64 pairs * PSTRIDE floats
    const int tid = threadIdx.x;

    // Stage W into LDS (K-pair interleaved), coalesced global reads.
    for (int i = tid; i < D * D; i += 256) {
        int k = i >> 7, c = i & 127;
        wlds[(k >> 1) * PSTRIDE + c * 2 + (k & 1)] = W[i];
    }
    __syncthreads();

    const int wave = tid >> 5;
    const int lane = tid & 31;
    const int half = lane >> 4;                // 0 = lanes 0-15, 1 = lanes 16-31
    const int l16  = lane & 15;

    const int rowBase = blockIdx.x * 128 + wave * 16;

    // A-fragment source row (clamped so OOB lanes read valid memory; their
    // results are discarded by the predicated store below).
    int arow = rowBase + l16;
    if (arow >= N_NODES) arow = N_NODES - 1;
    const float* aptr = x + (size_t)arow * D + half * 2;

    // Pull the back half of this row toward the caches while the first
    // K-steps run (global_prefetch_b8).
    __builtin_prefetch(aptr + 64, 0, 3);

    v8f acc[8];
    const v8f vzero = {0.f, 0.f, 0.f, 0.f, 0.f, 0.f, 0.f, 0.f};
#pragma unroll
    for (int nt = 0; nt < 8; ++nt) acc[nt] = vzero;

    for (int k0 = 0; k0 < D; k0 += 4) {
        v2f a = *reinterpret_cast<const v2f*>(aptr + k0);   // 8B aligned (k0 even)
        // Pair index: k0/2 for lanes 0-15, k0/2+1 for lanes 16-31.
        const float* bbase = wlds + ((k0 >> 1) + half) * PSTRIDE + l16 * 2;
#pragma unroll
        for (int nt = 0; nt < 8; ++nt) {
            v2f b = *reinterpret_cast<const v2f*>(bbase + nt * 32);  // ds_load_b64
            acc[nt] = __builtin_amdgcn_wmma_f32_16x16x4_f32(
                /*neg_a=*/false, a, /*neg_b=*/false, b,
                /*c_mod=*/(short)0, acc[nt], /*reuse_a=*/false, /*reuse_b=*/false);
        }
    }

    // Store D (16x16 per nt): VGPR i -> M = 8*half + i.
#pragma unroll
    for (int nt = 0; nt < 8; ++nt) {
#pragma unroll
        for (int i = 0; i < 8; ++i) {
            int row = rowBase + half * 8 + i;
            if (row < N_NODES)
                h[(size_t)row * D + nt * 16 + l16] = acc[nt][i];
        }
    }
}

// ---------------------------------------------------------------------------
// Kernel 4: edge scatter: agg[dst] += h[src] * (dis[src]*dis[dst])
//   One wave per edge; each lane handles 4 contiguous columns (float4 gather).
// ---------------------------------------------------------------------------
__global__ void gcn_edge_kernel(const int* __restrict__ src, const int* __restrict__ dst,
                                const float* __restrict__ dis,
                                const float* __restrict__ h,
                                float* __restrict__ agg) {
    int e    = blockIdx.x * 8 + (threadIdx.x >> 5);
    int lane = threadIdx.x & 31;
    if (e >= N_EDGES) return;

    int s = src[e];
    int d = dst[e];
    float norm = dis[s] * dis[d];

    const float4* hrow = reinterpret_cast<const float4*>(h + (size_t)s * D);
    float4 v = hrow[lane];

    float* arow = agg + (size_t)d * D + lane * 4;
    atomicAdd(arow + 0, v.x * norm);
    atomicAdd(arow + 1, v.y * norm);
    atomicAdd(arow + 2, v.z * norm);
    atomicAdd(arow + 3, v.w * norm);
}

// ---------------------------------------------------------------------------
// Kernel 5: fuse self-loop contribution + bias, accumulate column stats.
//   500 blocks x 200 rows; thread = (col, row-phase); 2 atomics per thread.
// ---------------------------------------------------------------------------
__global__ void gcn_stats_kernel(const float* __restrict__ h, const float* __restrict__ dis,
                                 const float* __restrict__ bias,
                                 float* __restrict__ agg,
                                 float* __restrict__ colsum, float* __restrict__ colsumsq) {
    int c   = threadIdx.x & 127;
    int sub = threadIdx.x >> 7;                // 0 or 1
    int r0  = blockIdx.x * 200;
    float bc = bias[c];

    float sum = 0.0f, sq = 0.0f;
    for (int r = r0 + sub; r < r0 + 200; r += 2) {
        float di = dis[r];
        size_t idx = (size_t)r * D + c;
        float v = agg[idx] + h[idx] * di * di + bc;   // self-loop norm = 1/deg
        agg[idx] = v;
        sum += v;
        sq  += v * v;
    }
    atomicAdd(&colsum[c], sum);
    atomicAdd(&colsumsq[c], sq);
}

// ---------------------------------------------------------------------------
// Kernel 6: BatchNorm (batch stats) + ReLU -> out
// ---------------------------------------------------------------------------
__global__ void gcn_bn_relu_kernel(const float* __restrict__ agg,
                                   const float* __restrict__ colsum,
                                   const float* __restrict__ colsumsq,
                                   const float* __restrict__ gamma,
                                   const float* __restrict__ beta,
                                   float* __restrict__ out) {
    size_t i = (size_t)blockIdx.x * 256 + threadIdx.x;
    if (i >= (size_t)N_NODES * D) return;
    int c = (int)(i & 127);
    const float invN = 1.0f / (float)N_NODES;
    float mean = colsum[c] * invN;
    float var  = colsumsq[c] * invN - mean * mean;
    float y = gamma[c] * (agg[i] - mean) * rsqrtf(var + BN_EPS) + beta[c];
    out[i] = fmaxf(y, 0.0f);
}

// ---------------------------------------------------------------------------
extern "C" void kernel_launch(void* const* d_in, const int* in_sizes, int n_in,
                              void* d_out, int out_size, void* d_ws, size_t ws_size,
                              hipStream_t stream) {
    const float* x     = (const float*)d_in[0];
    const float* W     = (const float*)d_in[1];
    const float* b     = (const float*)d_in[2];
    const float* gamma = (const float*)d_in[3];
    const float* beta  = (const float*)d_in[4];
    const int*   ei    = (const int*)d_in[5];      // [2, E] row-major
    const int*   src   = ei;
    const int*   dst   = ei + N_EDGES;

    float* ws       = (float*)d_ws;
    float* h        = ws;                                   // N*D
    float* agg      = h   + (size_t)N_NODES * D;            // N*D
    float* deg      = agg + (size_t)N_NODES * D;            // N  (deg -> deg^-1/2)
    float* colsum   = deg + N_NODES;                        // D
    float* colsumsq = colsum + D;                           // D
    float* out      = (float*)d_out;

    const dim3 blk(256);
    const size_t ND = (size_t)N_NODES * D;

    gcn_init_kernel <<<(unsigned)((ND + 255) / 256), blk, 0, stream>>>(agg, deg, colsum, colsumsq);
    gcn_deg_kernel  <<<(N_EDGES + 255) / 256, blk, 0, stream>>>(dst, deg);
    gcn_rsqrt_kernel<<<(N_NODES + 255) / 256, blk, 0, stream>>>(deg);
    gcn_gemm_wmma_kernel<<<(N_NODES + 127) / 128, blk, W_LDS_BYTES, stream>>>(x, W, h);
    gcn_edge_kernel <<<(N_EDGES + 7) / 8, blk, 0, stream>>>(src, dst, deg, h, agg);
    gcn_stats_kernel<<<500, blk, 0, stream>>>(h, deg, b, agg, colsum, colsumsq);
    gcn_bn_relu_kernel<<<(unsigned)((ND + 255) / 256), blk, 0, stream>>>(agg, colsum, colsumsq, gamma, beta, out);
}